// Generator_63660005261322
// MI455X (gfx1250) — compile-verified
//
#include <hip/hip_runtime.h>
#include <stdint.h>

typedef float v2f __attribute__((ext_vector_type(2)));
typedef float v8f __attribute__((ext_vector_type(8)));

#define T_STEPS 4096
#define BATCH   8192
#define NROWS   4095          // dW rows
#define NSEG    8
#define SEG_ROWS 512
#define NBLK    32            // 16-row blocks per segment
#define OS_STR  36            // outStage batch stride in floats (144B: 16B aligned, bank-conflict-free)

__device__ __forceinline__ uint32_t lds_off(const void* p) {
  return (uint32_t)(uintptr_t)p;
}
__device__ __forceinline__ void async_load_b128(uint32_t lds, const float* g) {
  asm volatile("global_load_async_to_lds_b128 %0, %1, off"
               :: "v"(lds), "v"(g) : "memory");
}
__device__ __forceinline__ void async_store_b128(const float* g, uint32_t lds) {
  asm volatile("global_store_async_from_lds_b128 %0, %1, off"
               :: "v"(g), "v"(lds) : "memory");
}
__device__ __forceinline__ void wait_async0() {
  asm volatile("s_wait_asynccnt 0x0" ::: "memory");
}
__device__ __forceinline__ void wait_ds0() {
  asm volatile("s_wait_dscnt 0x0" ::: "memory");
}

// -------- Pass 1: per-segment per-batch totals of dW --------------------
__global__ __launch_bounds__(256) void seg_reduce(const float* __restrict__ dW,
                                                  float* __restrict__ segTot) {
  int tid = blockIdx.x * 256 + threadIdx.x;   // 65536 threads
  int b = tid & (BATCH - 1);
  int s = tid >> 13;
  int base = s * SEG_ROWS;
  int nrow = (s == NSEG - 1) ? (SEG_ROWS - 1) : SEG_ROWS;  // last segment: 511 valid rows
  const float* p = dW + (size_t)base * BATCH + b;
  float acc = 0.0f;
#pragma unroll 8
  for (int r = 0; r < nrow; ++r) acc += p[(size_t)r * BATCH];
  segTot[s * BATCH + b] = acc;
}

// -------- Pass 2: WMMA blocked scan + async emit ------------------------
// One wave (32 threads) per (16-batch group, segment). 512*8 = 4096 blocks.
__global__ __launch_bounds__(32) void sde_scan(const float* __restrict__ dW,
                                               const float* __restrict__ segTot,
                                               const float* __restrict__ driftp,
                                               const float* __restrict__ diffp,
                                               float* __restrict__ out) {
  __shared__ float tileIn[2][16 * 16];      // [row t][batch n], double-buffered
  __shared__ float outStage[16 * OS_STR];   // [batch n][pair i][2], padded stride

  const int l = threadIdx.x;
  const int n = l & 15;                     // M index (A) / N index (B,C,D)
  const int h = l >> 4;                     // lane half
  const int g = blockIdx.x >> 3;            // batch group
  const int s = blockIdx.x & 7;             // segment
  const int b0 = g * 16;
  const float drift = driftp[0];
  const float diffusion = diffp[0];

  // A = lower-triangular ones, split into 4 K-chunks of 16x4.
  // A layout: lane l holds M = l&15; vgpr v holds K = 4j + v + 2*(l>=16).
  v2f A[4];
#pragma unroll
  for (int j = 0; j < 4; ++j) {
    A[j].x = (4 * j + 2 * h + 0 <= n) ? 1.0f : 0.0f;
    A[j].y = (4 * j + 2 * h + 1 <= n) ? 1.0f : 0.0f;
  }

  // Initial carry = sum of earlier segments' totals for batch b0 + (l&15).
  // This per-lane pattern matches the C/D matrix column layout exactly.
  float carry = 0.0f;
  for (int sp = 0; sp < s; ++sp) carry += segTot[sp * BATCH + b0 + n];

  const int tseg = s * SEG_ROWS;
  const int row_lo = l >> 2;                // 0..7
  const int c4 = (l & 3) * 4;               // 16B segment within 64B row

  // Incremental per-lane global source pointers (rows row_lo and 8+row_lo of each tile).
  const float* pa = dW + (size_t)(tseg + row_lo) * BATCH + b0 + c4;
  const float* pb = pa + (size_t)8 * BATCH;
  // Only segment 7, tile 31, lane-row 15 would touch nonexistent dW row 4095; it feeds
  // only the unused D[15], so just re-read row 4094 there.
  const bool fixlane = (s == NSEG - 1) && (row_lo == 7);

  // LDS targets for the two async loads, per buffer.
  const uint32_t tinA[2] = { lds_off(&tileIn[0][row_lo * 16 + c4]),
                             lds_off(&tileIn[1][row_lo * 16 + c4]) };
  const uint32_t tinB[2] = { lds_off(&tileIn[0][(8 + row_lo) * 16 + c4]),
                             lds_off(&tileIn[1][(8 + row_lo) * 16 + c4]) };

  // Prologue: prefetch tile 0 (never contains row 4095).
  async_load_b128(tinA[0], pa);
  async_load_b128(tinB[0], pb);
  pa += (size_t)16 * BATCH;
  pb += (size_t)16 * BATCH;

  // Incremental output pointer + fixed LDS source offsets for the emit phase.
  const int bb = l >> 1;
  const float* pout = out + ((size_t)(b0 + bb) * T_STEPS + tseg) * 2 + (l & 1) * 4;
  uint32_t osOff[4];
#pragma unroll
  for (int k = 0; k < 4; ++k)
    osOff[k] = lds_off(&outStage[bb * OS_STR + ((l & 1) + 2 * k) * 4]);

  // Float time base: tau for slot r (r=0..7) is tbase + hbase + r.
  float tbase = (float)tseg;
  const float hbase = (float)(8 * h);

  int cur = 0;
  for (int blk = 0; blk < NBLK; ++blk) {
    wait_async0();  // prefetched tile ready; previous outStage async stores drained

    // Prefetch next tile into the other buffer (overlaps WMMA below).
    if (blk + 1 < NBLK) {
      async_load_b128(tinA[cur ^ 1], pa);
      const float* pbUse = (fixlane && blk == NBLK - 2) ? (pb - BATCH) : pb;
      async_load_b128(tinB[cur ^ 1], pbUse);
      pa += (size_t)16 * BATCH;
      pb += (size_t)16 * BATCH;
    }

    // Preload all four B operands (4x ds_load_2addr pipelined under one wait),
    // then run the chained WMMA accumulation D = L*X + carry.
    v2f bm[4];
#pragma unroll
    for (int j = 0; j < 4; ++j) {
      bm[j].x = tileIn[cur][(4 * j + 2 * h) * 16 + n];
      bm[j].y = tileIn[cur][(4 * j + 2 * h + 1) * 16 + n];
    }
    v8f d = { carry, carry, carry, carry, carry, carry, carry, carry };
#pragma unroll
    for (int j = 0; j < 4; ++j) {
      d = __builtin_amdgcn_wmma_f32_16x16x4_f32(
              false, A[j], false, bm[j], (short)0, d, false, false);
    }

    // Row 7 (i=8 emission on h=1 lanes) and row 15 (carry-out) both live in vgpr 7.
    float d7 = d[7];
    float row7  = __int_as_float(__builtin_amdgcn_ds_bpermute((n) << 2,      __float_as_int(d7)));
    float row15 = __int_as_float(__builtin_amdgcn_ds_bpermute((16 + n) << 2, __float_as_int(d7)));

    // Emit window tau = t0 .. t0+15 (128B aligned in out):
    //   i = 0  -> x = carry_in  (for t0==0 this yields x0 = 0)
    //   i = k  -> x = D[k-1]
    {
      const float tb2 = tbase + hbase;
      float v0 = h ? row7 : carry;
      float2 p0;
      p0.x = tb2;
      p0.y = diffusion * v0 + drift * tb2;
      *(float2*)&outStage[n * OS_STR + (8 * h) * 2] = p0;
#pragma unroll
      for (int r = 1; r < 8; ++r) {
        float tau = tb2 + (float)r;
        float2 p;
        p.x = tau;
        p.y = diffusion * d[r - 1] + drift * tau;
        *(float2*)&outStage[n * OS_STR + (8 * h + r) * 2] = p;
      }
    }
    wait_ds0();  // LDS staging visible before async engine reads it

    // Coalesced emit: per batch, 16 (tau,x) pairs = 128B contiguous.
#pragma unroll
    for (int k = 0; k < 4; ++k)
      async_store_b128(pout + 8 * k, osOff[k]);
    pout += 32;

    carry = row15;
    tbase += 16.0f;
    cur ^= 1;
  }
  wait_async0();  // drain final stores before wave exit
}

// ------------------------------------------------------------------------
extern "C" void kernel_launch(void* const* d_in, const int* in_sizes, int n_in,
                              void* d_out, int out_size, void* d_ws, size_t ws_size,
                              hipStream_t stream) {
  // inputs: [0]=ts(4096 f32, == arange, unused), [1]=dW(4095*8192 f32),
  //         [2]=drift(1 f32), [3]=diffusion(1 f32), [4]=batch_size (unused)
  const float* dW   = (const float*)d_in[1];
  const float* drft = (const float*)d_in[2];
  const float* diff = (const float*)d_in[3];
  float* out = (float*)d_out;
  float* segTot = (float*)d_ws;             // NSEG*BATCH floats = 256KB scratch

  seg_reduce<<<(NSEG * BATCH) / 256, 256, 0, stream>>>(dW, segTot);
  sde_scan<<<(BATCH / 16) * NSEG, 32, 0, stream>>>(dW, segTot, drft, diff, out);
}